// LinearCondensed_81260781240864
// MI455X (gfx1250) — compile-verified
//
#include <hip/hip_runtime.h>
#include <stdint.h>

// Problem constants (from reference)
#define B_DIM   1024
#define IN_LEN  4096
#define OUT_DIM 4096
#define FAN     128

// Tiling
#define B_TILE   8              // batch rows staged in LDS per workgroup
#define STRIDE   10             // dwords per j-row in LDS: 8 data + 2 pad (even -> float2 aligned)
#define NTHREADS 256            // 8 waves (wave32)
#define O_SPLITS 4
#define O_PER_WG (OUT_DIM / O_SPLITS)        // 1024
#define WAVES    (NTHREADS / 32)             // 8
#define O_PER_WAVE (O_PER_WG / WAVES)        // 128

// ---- CDNA5 async global->LDS path -----------------------------------------
#if defined(__has_builtin)
#  if __has_builtin(__builtin_amdgcn_global_load_async_to_lds_b32)
#    define HAVE_ASYNC_BUILTIN 1
#  endif
#  if __has_builtin(__builtin_amdgcn_s_wait_asynccnt)
#    define HAVE_ASYNCCNT_BUILTIN 1
#  endif
#endif

typedef __attribute__((address_space(1))) int* gint_ptr;
typedef __attribute__((address_space(3))) int* lint_ptr;

__device__ __forceinline__ void async_copy_b32(const float* gsrc, float* lds_dst) {
#if defined(HAVE_ASYNC_BUILTIN)
    __builtin_amdgcn_global_load_async_to_lds_b32(
        (gint_ptr)(uintptr_t)gsrc,
        (lint_ptr)(uintptr_t)lds_dst,
        0, 0);
#else
    // inline-asm fallback: vdst = LDS byte address, vaddr = 64-bit global address
    unsigned lds_off = (unsigned)(uintptr_t)lds_dst;
    asm volatile("global_load_async_to_lds_b32 %0, %1, off"
                 :
                 : "v"(lds_off), "v"(gsrc)
                 : "memory");
#endif
}

__device__ __forceinline__ void wait_async_zero() {
#if defined(HAVE_ASYNCCNT_BUILTIN)
    __builtin_amdgcn_s_wait_asynccnt(0);
#else
    asm volatile("s_wait_asynccnt 0" ::: "memory");
#endif
}
// ---------------------------------------------------------------------------

__global__ __launch_bounds__(NTHREADS)
void lincond_kernel(const float* __restrict__ input,
                    const float* __restrict__ weight,
                    const float* __restrict__ bias,
                    const int*   __restrict__ idx,
                    float*       __restrict__ out)
{
    __shared__ float Xt[IN_LEN * STRIDE];   // 4096 * 10 * 4B = 160 KB -> 2 WGs / WGP

    const int tid = threadIdx.x;
    const int b0  = blockIdx.x * B_TILE;            // batch tile origin
    const int o_wg = blockIdx.y * O_PER_WG;         // output range origin

    // ---- Phase 1: async-load input tile into LDS, transposed Xt[j][b] ----
    // e = i*256 + tid : j consecutive within an i (coalesced global reads),
    // LDS banks (10*tid + b) mod 64 -> 32 distinct banks per wave.
    {
        const float* src = input + (size_t)b0 * IN_LEN;
        #pragma unroll 4
        for (int i = 0; i < (B_TILE * IN_LEN) / NTHREADS; ++i) {
            int e = i * NTHREADS + tid;
            int j = e & (IN_LEN - 1);
            int b = e >> 12;                        // IN_LEN == 4096
            async_copy_b32(src + e, &Xt[j * STRIDE + b]);
        }
    }
    wait_async_zero();
    __syncthreads();

    // ---- Phase 2: gather-dot ----
    // lane layout: o_group = lane>>2 (8 outputs per wave step), b_pair = lane&3
    // One ds_load_b64 per lane = 2 batch elements; wave = 64 gathered operands,
    // the b64 minimum for 512M total gathers. idx is shared across all b, so the
    // gather cost is amortized across the batch tile.
    const int wave = tid >> 5;
    const int lane = tid & 31;
    const int og   = lane >> 2;                     // 0..7
    const int bp   = lane & 3;                      // 0..3  (b = 2bp, 2bp+1)
    const int o_wave = o_wg + wave * O_PER_WAVE;

    for (int c = 0; c < O_PER_WAVE / 8; ++c) {
        const int o = o_wave + c * 8 + og;
        const int*   ip = idx    + (size_t)o * FAN;
        const float* wp = weight + (size_t)o * FAN;

        // prefetch next chunk's index row into L2 (CDNA5 global_prefetch_b8)
        __builtin_prefetch(ip + 8 * FAN, 0, 1);

        float2 acc = make_float2(0.f, 0.f);
        #pragma unroll 4
        for (int f = 0; f < FAN; f += 4) {
            int4   j4 = *(const int4*)  (ip + f);   // 4 gather indices
            float4 w4 = *(const float4*)(wp + f);   // 4 weights

            float2 x0 = *(const float2*)&Xt[j4.x * STRIDE + 2 * bp];
            float2 x1 = *(const float2*)&Xt[j4.y * STRIDE + 2 * bp];
            float2 x2 = *(const float2*)&Xt[j4.z * STRIDE + 2 * bp];
            float2 x3 = *(const float2*)&Xt[j4.w * STRIDE + 2 * bp];

            acc.x = fmaf(w4.x, x0.x, acc.x);  acc.y = fmaf(w4.x, x0.y, acc.y);
            acc.x = fmaf(w4.y, x1.x, acc.x);  acc.y = fmaf(w4.y, x1.y, acc.y);
            acc.x = fmaf(w4.z, x2.x, acc.x);  acc.y = fmaf(w4.z, x2.y, acc.y);
            acc.x = fmaf(w4.w, x3.x, acc.x);  acc.y = fmaf(w4.w, x3.y, acc.y);
        }

        // out is written once and never re-read: non-temporal stores keep the
        // streaming 16 MB out of L2 so input/idx/weight stay resident.
        const float bv = bias[o];
        const int b = b0 + 2 * bp;
        __builtin_nontemporal_store(acc.x + bv, &out[(size_t)b       * OUT_DIM + o]);
        __builtin_nontemporal_store(acc.y + bv, &out[(size_t)(b + 1) * OUT_DIM + o]);
    }
}

extern "C" void kernel_launch(void* const* d_in, const int* in_sizes, int n_in,
                              void* d_out, int out_size, void* d_ws, size_t ws_size,
                              hipStream_t stream) {
    (void)in_sizes; (void)n_in; (void)out_size; (void)d_ws; (void)ws_size;
    const float* input  = (const float*)d_in[0];   // (B, IN_LEN) f32
    const float* weight = (const float*)d_in[1];   // (OUT, FAN) f32
    const float* bias   = (const float*)d_in[2];   // (OUT,) f32
    const int*   idx    = (const int*)  d_in[3];   // (OUT, FAN) i32
    float* out = (float*)d_out;                    // (B, OUT) f32

    dim3 grid(B_DIM / B_TILE, O_SPLITS);           // 128 x 4 = 512 workgroups
    lincond_kernel<<<grid, NTHREADS, 0, stream>>>(input, weight, bias, idx, out);
}